// Model_NoSM_ControlA_65850438582600
// MI455X (gfx1250) — compile-verified
//
#include <hip/hip_runtime.h>
#include <math.h>

#define NTC 2048
#define BBC 4096
#define SPLITS 4
#define KLEN (NTC / SPLITS)   // 512
#define T_CONTROL_C 3.0f
#define TWO_PI_F 6.28318530717958647692f

typedef __attribute__((ext_vector_type(2))) float v2f;
typedef __attribute__((ext_vector_type(8))) float v8f;

// ---- ws param block layout, per dataset (stride 32 floats) ----
// 0 alpha, 1 g2, 2 i00, 3 i11, 4 i12, 5 i22, 6 det_m,
// 7 s00, 8 s01, 9 s11, 10 det_s, 11 q, 12 W_aa, 13 sig_a2,
// 14 sum(wf^2), 15 sum(wf*wv), 16 sum(wv^2), 17 Tm

// ---------------- Kernel 1: Tm = T_control - mean(t_wait) ----------------
__global__ void k_twait(const float* __restrict__ tw, const float* __restrict__ twSM,
                        float* __restrict__ P) {
  __shared__ float r0[256], r1[256];
  int tid = threadIdx.x;
  float s0 = 0.f, s1 = 0.f;
  for (int i = tid; i < BBC; i += 256) { s0 += tw[i]; s1 += twSM[i]; }
  r0[tid] = s0; r1[tid] = s1; __syncthreads();
  for (int s = 128; s > 0; s >>= 1) {
    if (tid < s) { r0[tid] += r0[tid + s]; r1[tid] += r1[tid + s]; }
    __syncthreads();
  }
  if (tid == 0) {
    P[17]      = T_CONTROL_C - r0[0] / (float)BBC;
    P[32 + 17] = T_CONTROL_C - r1[0] / (float)BBC;
  }
}

// ---------------- scalar Sigma / threshold math (t_obs = 1) ----------------
__device__ void sigma_scalars(float sig_f, float sig_a, float xi_f, float xi_v,
                              float d0, float* o) {
  const float t_avg = 0.5f, t2 = 1.f / 3.f, Ct = 1.f / 12.f;
  float xf = xi_f / sig_f, xv = xi_v / sig_f;
  float ixf2 = 1.f / (xf * xf), ixv2 = 1.f / (xv * xv);
  float det_inv = (1.f + ixf2) * (Ct + ixv2) + t_avg * t_avg * ixf2;
  float pre = sig_f * sig_f / (det_inv * det_inv);
  float Pff = pre * ((Ct + ixv2) * (Ct + ixv2) + Ct * t_avg * t_avg);
  float Pfv = pre * t_avg * (ixf2 * ixv2 - Ct);
  float Pvv = pre * (t_avg * t_avg * ixf2 * ixf2 + Ct * (1.f + ixf2) * (1.f + ixf2));
  float detP = Pff * Pvv - Pfv * Pfv;
  float sa2 = sig_a * sig_a;
  o[0] = 1.f / sa2;        // i00 (Sinv_m[0,0])
  o[1] = Pvv / detP;       // i11
  o[2] = -Pfv / detP;      // i12
  o[3] = Pff / detP;       // i22
  o[4] = sa2 * detP;       // det_m
  float pre2 = sig_f * sig_f;
  float Qff = pre2 / ((1.f + ixf2) * (1.f + ixf2));
  float Qfv = pre2 * (t_avg / (xf * xf + 1.f)) / det_inv;
  float Qvv = pre2 * (t_avg * t_avg * ixf2 * ixf2 + Ct * (1.f + ixf2) * (1.f + ixf2)) /
              (det_inv * det_inv);
  float detQ = Qff * Qvv - Qfv * Qfv;
  o[5] = Qvv / detQ;   // s00
  o[6] = -Qfv / detQ;  // s01
  o[7] = Qff / detQ;   // s11
  o[8] = detQ;         // det_s
  float log_arg = 1.f + xv * xv * (Ct + t2 / (1.f + xf * xf));
  float det_ratio = (1.f / (sig_f * sig_f)) * (Ct + ixv2 + t2 / (1.f + xf * xf));
  float tsq = (logf(log_arg) - 2.f * d0) / det_ratio;
  o[9] = sqrtf(fmaxf(tsq, 0.f));  // q
}

// ---------------- Kernel 2: control weights + all constants ----------------
__global__ void k_weights(const float* __restrict__ t, const float* __restrict__ tSM,
                          const float* hsig_f, const float* hsig_w, const float* hxi_f,
                          const float* hxi_w, const float* hxi_v, const float* delta0,
                          const float* halpha, const float* hg, const float* hW_aa,
                          const float* hsig_a,
                          float* __restrict__ P, float* __restrict__ wf1,
                          float* __restrict__ wv1, float* __restrict__ wf2,
                          float* __restrict__ wv2) {
  __shared__ float red[6][256];
  int tid = threadIdx.x;
  float alpha = expf(halpha[0]);
  float W_aa = expf(hW_aa[0]);
  float alpha2 = alpha * W_aa;
  float Tm1 = P[17], Tm2 = P[32 + 17];
  float a1 = 0, b1 = 0, c1 = 0, a2 = 0, b2 = 0, c2 = 0;
  for (int i = tid; i < NTC; i += 256) {
    float rem = fmaxf(Tm1 - t[i], 0.f);                 // tau = t[0] row
    float wfA = alpha * rem / Tm1;
    float wvA = alpha * rem * rem / (2.f * Tm1 * Tm1);
    wf1[i] = wfA; wv1[i] = wvA;
    a1 += wfA * wfA; b1 += wfA * wvA; c1 += wvA * wvA;
    float rem2 = fmaxf(Tm2 - tSM[i], 0.f);              // tau = t_SM[0] row
    float wfB = alpha2 * rem2 / Tm2;
    float wvB = alpha2 * rem2 * rem2 / (2.f * Tm2 * Tm2);
    wf2[i] = wfB; wv2[i] = wvB;
    a2 += wfB * wfB; b2 += wfB * wvB; c2 += wvB * wvB;
  }
  red[0][tid] = a1; red[1][tid] = b1; red[2][tid] = c1;
  red[3][tid] = a2; red[4][tid] = b2; red[5][tid] = c2;
  __syncthreads();
  for (int s = 128; s > 0; s >>= 1) {
    if (tid < s)
      for (int j = 0; j < 6; ++j) red[j][tid] += red[j][tid + s];
    __syncthreads();
  }
  if (tid == 0) {
    float sig_f = expf(hsig_f[0]);
    float sig_w = expf(hsig_w[0]);
    float xi_f = expf(hxi_f[0]);
    float xi_w = expf(hxi_w[0]);
    float xi_v = expf(hxi_v[0]);
    float d0 = delta0[0];
    float g = expf(hg[0]);
    float sig_a = expf(hsig_a[0]);
    float o[10];
    // dataset 1
    sigma_scalars(sig_f, sig_a, xi_f, xi_v, d0, o);
    P[0] = alpha; P[1] = g * g;
    P[2] = o[0]; P[3] = o[1]; P[4] = o[2]; P[5] = o[3]; P[6] = o[4];
    P[7] = o[5]; P[8] = o[6]; P[9] = o[7]; P[10] = o[8]; P[11] = o[9];
    P[12] = W_aa; P[13] = sig_a * sig_a;
    P[14] = red[0][0]; P[15] = red[1][0]; P[16] = red[2][0];
    // dataset 2 (SM)
    float sig_eff = sqrtf(sig_f * sig_f + sig_w * sig_w);
    float xi_ve = sqrtf(xi_v * xi_v + xi_w * xi_w);
    sigma_scalars(sig_eff, sig_a, xi_f, xi_ve, d0, o);
    float* Q = P + 32;
    Q[0] = alpha2; Q[1] = g * g;
    Q[2] = o[0]; Q[3] = o[1]; Q[4] = o[2]; Q[5] = o[3]; Q[6] = o[4];
    Q[7] = o[5]; Q[8] = o[6]; Q[9] = o[7]; Q[10] = o[8]; Q[11] = o[9];
    Q[12] = W_aa; Q[13] = sig_a * sig_a;
    Q[14] = red[3][0]; Q[15] = red[4][0]; Q[16] = red[5][0];
  }
}

// ---------------- Kernel 3: WMMA streaming reductions (split-K) ----------------
// One wave handles 16 batch rows over KLEN time steps (split blockIdx.y).
// Per 8-step chunk each lane loads one float4 of u,x,wf,wv at time
// k0 + 4*hi (hi = lane/16).  K-slot -> time mapping per WMMA:
//   WMMA#1: slots {0,1,2,3} -> times {k0, k0+1, k0+4, k0+5}
//   WMMA#2: slots {0,1,2,3} -> times {k0+2, k0+3, k0+6, k0+7}
// A and B fragments use the same mapping, so sums over k are exact.
// Weighted WMMA -> columns 0/1 accumulate S1,S2; Gram WMMA (B := A) ->
// diagonal accumulates S0.  2-stage software pipeline keeps loads in flight.
__global__ __launch_bounds__(128) void k_rows(
    const float* __restrict__ u, const float* __restrict__ x,
    const float* __restrict__ wf, const float* __restrict__ wv,
    const float* __restrict__ Pds, float* __restrict__ S0,
    float* __restrict__ S1, float* __restrict__ S2) {
  __shared__ float ldsW[4][16][16];
  __shared__ float ldsG[4][16][16];
  int lane = threadIdx.x & 31;
  int wave = threadIdx.x >> 5;
  int m = lane & 15, hi = lane >> 4;
  int row0 = (blockIdx.x * 4 + wave) * 16;
  int split = blockIdx.y;
  int kbase = split * KLEN;
  float alpha = Pds[0];
  const float* up = u + (size_t)(row0 + m) * NTC + kbase + 4 * hi;
  const float* xp = x + (size_t)(row0 + m) * NTC + kbase + 4 * hi;
  const float* fp = wf + kbase + 4 * hi;
  const float* vp = wv + kbase + 4 * hi;
  v8f accW = {};
  v8f accG = {};
  float4 ucur = *(const float4*)(up);
  float4 xcur = *(const float4*)(xp);
  float4 fcur = *(const float4*)(fp);
  float4 vcur = *(const float4*)(vp);
  for (int k0 = 0; k0 < KLEN; k0 += 8) {
    int kn = (k0 + 8 < KLEN) ? (k0 + 8) : k0;  // last iter: dummy reload
    float4 unxt = *(const float4*)(up + kn);
    float4 xnxt = *(const float4*)(xp + kn);
    float4 fnxt = *(const float4*)(fp + kn);
    float4 vnxt = *(const float4*)(vp + kn);
    __builtin_prefetch(up + k0 + 1024, 0, 0);
    __builtin_prefetch(xp + k0 + 1024, 0, 0);
    // chunk lo: times k0+4hi+{0,1}
    v2f a1, b1;
    a1.x = fmaf(-alpha, xcur.x, ucur.x);
    a1.y = fmaf(-alpha, xcur.y, ucur.y);
    b1.x = (m == 0) ? fcur.x : ((m == 1) ? vcur.x : 0.f);
    b1.y = (m == 0) ? fcur.y : ((m == 1) ? vcur.y : 0.f);
    accG = __builtin_amdgcn_wmma_f32_16x16x4_f32(false, a1, false, a1, (short)0,
                                                 accG, false, false);
    accW = __builtin_amdgcn_wmma_f32_16x16x4_f32(false, a1, false, b1, (short)0,
                                                 accW, false, false);
    // chunk hi: times k0+4hi+{2,3}
    v2f a2, b2;
    a2.x = fmaf(-alpha, xcur.z, ucur.z);
    a2.y = fmaf(-alpha, xcur.w, ucur.w);
    b2.x = (m == 0) ? fcur.z : ((m == 1) ? vcur.z : 0.f);
    b2.y = (m == 0) ? fcur.w : ((m == 1) ? vcur.w : 0.f);
    accG = __builtin_amdgcn_wmma_f32_16x16x4_f32(false, a2, false, a2, (short)0,
                                                 accG, false, false);
    accW = __builtin_amdgcn_wmma_f32_16x16x4_f32(false, a2, false, b2, (short)0,
                                                 accW, false, false);
    ucur = unxt; xcur = xnxt; fcur = fnxt; vcur = vnxt;
  }
  // C/D layout: lane, VGPR v -> (M = v + 8*hi, N = lane%16)
#pragma unroll
  for (int v = 0; v < 8; ++v) {
    ldsW[wave][v + 8 * hi][m] = accW[v];
    ldsG[wave][v + 8 * hi][m] = accG[v];
  }
  __syncthreads();
  if (lane < 16) {
    int r = lane;
    S0[split * BBC + row0 + r] = ldsG[wave][r][r];
    S1[split * BBC + row0 + r] = ldsW[wave][r][0];
    S2[split * BBC + row0 + r] = ldsW[wave][r][1];
  }
}

// ---------------- Kernel 4: per-row likelihood finalization ----------------
__global__ void k_final(const float* __restrict__ z, const float* __restrict__ zSM,
                        const float* __restrict__ t, const float* __restrict__ tSM,
                        const float* __restrict__ P, const float* __restrict__ S,
                        float* __restrict__ partial) {
  __shared__ float red[256];
  int b = blockIdx.x * 256 + threadIdx.x;
  float total = 0.f;
  for (int ds = 0; ds < 2; ++ds) {
    const float* Pp = P + ds * 32;
    float alpha = Pp[0], g2 = Pp[1];
    float i00 = Pp[2], i11 = Pp[3], i12 = Pp[4], i22 = Pp[5], det_m = Pp[6];
    float s00 = Pp[7], s01 = Pp[8], s11 = Pp[9], det_s = Pp[10];
    float q = Pp[11], W_aa = Pp[12], sa2 = Pp[13];
    float sff = Pp[14], sfv = Pp[15], svv = Pp[16];
    const float* zz = ds ? zSM : z;
    float z0 = zz[b * 3 + 0], z1 = zz[b * 3 + 1], z2 = zz[b * 3 + 2];
    // dt = t[:,1]-t[:,0]; dt_SM = t_SM[:,1]-t[:,0] (reference uses t[:,0] both)
    float dt = (ds ? tSM[(size_t)b * NTC + 1] : t[(size_t)b * NTC + 1]) -
               t[(size_t)b * NTC];
    const float* Sd = S + ds * 3 * SPLITS * BBC;
    float S0b = 0.f, S1b = 0.f, S2b = 0.f;
#pragma unroll
    for (int s = 0; s < SPLITS; ++s) {
      S0b += Sd[s * BBC + b];
      S1b += Sd[(SPLITS + s) * BBC + b];
      S2b += Sd[(2 * SPLITS + s) * BBC + b];
    }
    float r = dt / g2;
    // ---- move branch (3x3) ----
    float A00 = i00 + r * sff, A01 = r * sfv, A11 = i11 + r * svv;
    float detA = A00 * A11 - A01 * A01;
    float inv00 = A11 / detA, inv01 = -A01 / detA, inv11 = A00 / detA;
    float Woz0 = W_aa * z0, Woz1 = z1, Wvz = alpha * z2;
    float Jr0 = Woz0 * i00 + r * S1b;
    float Jr1 = Woz1 * i11 + Wvz * i12 + r * S2b;
    float Jv0 = r * sff;
    float Jv1 = i12 + r * sfv;  // w_tv = w_f
    float AiJv0 = inv00 * Jv0 + inv01 * Jv1;
    float AiJv1 = inv01 * Jv0 + inv11 * Jv1;
    float JrAiJv = Jr0 * AiJv0 + Jr1 * AiJv1;
    float JvAiJv = Jv0 * AiJv0 + Jv1 * AiJv1;
    float AiJr0 = inv00 * Jr0 + inv01 * Jr1;
    float AiJr1 = inv01 * Jr0 + inv11 * Jr1;
    float JrAiJr = Jr0 * AiJr0 + Jr1 * AiJr1;
    float J = i22 * Wvz + Woz1 * i12 + r * S1b - JrAiJv;
    float kk = i22 + r * sff - JvAiJv;
    float quadW = 0.5f * (W_aa * W_aa * i00 * z0 * z0 + i11 * z1 * z1 +
                          2.f * alpha * i12 * z1 * z2 + alpha * alpha * i22 * z2 * z2);
    float Zm = quadW + 0.5f * r * S0b - 0.5f * JrAiJr - 0.5f * J * J / kk;
    float pre_m = kk * detA * det_m * sa2;
    float sq = sqrtf(2.f * kk);
    float erfm = 1.f + 0.5f * erff((J - kk * q) / sq) - 0.5f * erff((J + kk * q) / sq);
    float Lm = expf(-Zm) * erfm * rsqrtf(pre_m);
    // ---- stat branch (2x2) ----
    float As = s00 + r * sff;
    float Wvzs = alpha * z2;
    float Jrs = z1 * s00 + Wvzs * s01 + r * S1b;
    float Js = s11 * Wvzs + z1 * s01 - Jrs * s01 / As;
    float ks = s11 - s01 * s01 / As;
    float quadWs = 0.5f * (s00 * z1 * z1 + 2.f * alpha * s01 * z1 * z2 +
                           alpha * alpha * s11 * z2 * z2);
    float Zs = quadWs + 0.5f * r * S0b - 0.5f * Jrs * Jrs / As - 0.5f * Js * Js / ks;
    float pre_s = ks * As * det_s;
    float sqs = sqrtf(2.f * ks);
    float erfs = 0.5f * erff((Js + ks * q) / sqs) - 0.5f * erff((Js - ks * q) / sqs);
    float Ls = expf(-Zs) * erfs * rsqrtf(pre_s);
    total += (float)NTC * logf(TWO_PI_F * g2 / dt) - logf(Lm + Ls);
  }
  red[threadIdx.x] = total;
  __syncthreads();
  for (int s = 128; s > 0; s >>= 1) {
    if (threadIdx.x < s) red[threadIdx.x] += red[threadIdx.x + s];
    __syncthreads();
  }
  if (threadIdx.x == 0) partial[blockIdx.x] = red[0];
}

// ---------------- Kernel 5: final scalar ----------------
__global__ void k_sum(const float* __restrict__ partial, float* __restrict__ out) {
  float s = 0.f;
  for (int i = 0; i < 16; ++i) s += partial[i];
  out[0] = s;
}

extern "C" void kernel_launch(void* const* d_in, const int* in_sizes, int n_in,
                              void* d_out, int out_size, void* d_ws, size_t ws_size,
                              hipStream_t stream) {
  const float* u_t   = (const float*)d_in[0];
  const float* x_t   = (const float*)d_in[1];
  const float* z     = (const float*)d_in[2];
  const float* t     = (const float*)d_in[3];
  const float* tw    = (const float*)d_in[4];
  const float* u_SM  = (const float*)d_in[5];
  const float* x_SM  = (const float*)d_in[6];
  const float* z_SM  = (const float*)d_in[7];
  const float* t_SM  = (const float*)d_in[8];
  const float* tw_SM = (const float*)d_in[9];

  float* ws = (float*)d_ws;
  float* P   = ws;                  // 64 floats
  float* wf1 = ws + 64;             // NT
  float* wv1 = wf1 + NTC;
  float* wf2 = wv1 + NTC;
  float* wv2 = wf2 + NTC;
  float* S   = ws + 64 + 4 * NTC;   // 2 * 3 * SPLITS * B floats
  float* partial = S + 2 * 3 * SPLITS * BBC;  // 16 floats

  k_twait<<<1, 256, 0, stream>>>(tw, tw_SM, P);
  k_weights<<<1, 256, 0, stream>>>(t, t_SM,
      (const float*)d_in[10], (const float*)d_in[11], (const float*)d_in[12],
      (const float*)d_in[13], (const float*)d_in[14], (const float*)d_in[15],
      (const float*)d_in[16], (const float*)d_in[17], (const float*)d_in[18],
      (const float*)d_in[19], P, wf1, wv1, wf2, wv2);

  dim3 grid(BBC / 64, SPLITS);  // 4 waves/block * 16 rows/wave; split-K in y
  float* S1d = S;
  float* S2d = S + 3 * SPLITS * BBC;
  k_rows<<<grid, 128, 0, stream>>>(u_t, x_t, wf1, wv1, P,
                                   S1d, S1d + SPLITS * BBC, S1d + 2 * SPLITS * BBC);
  k_rows<<<grid, 128, 0, stream>>>(u_SM, x_SM, wf2, wv2, P + 32,
                                   S2d, S2d + SPLITS * BBC, S2d + 2 * SPLITS * BBC);

  k_final<<<BBC / 256, 256, 0, stream>>>(z, z_SM, t, t_SM, P, S, partial);
  k_sum<<<1, 1, 0, stream>>>(partial, (float*)d_out);
}